// GridAttentionMap_64364379898403
// MI455X (gfx1250) — compile-verified
//
#include <hip/hip_runtime.h>

// ---------------------------------------------------------------------------
// Criss-cross axial attention (n=1, C=256, t=8, h=w=64, heads=8, S=134)
//   pv  = conv_w @ value + conv_b            (GEMM 256x32768x256, WMMA f32)
//   out = t-axis + h-axis + w-axis aggregation of pv with attention weights
// Memory-bound (~240 MB moved / 23.3 TB/s ~= 10 us), so fp32 WMMA
// (V_WMMA_F32_16X16X4_F32) keeps exact precision at no cost.
// Axis passes stage pv tile + attention slice in LDS (padded stride 65,
// bank-conflict-free) so WMMA operands come from ds_load, not strided VMEM.
// ---------------------------------------------------------------------------

typedef __attribute__((ext_vector_type(2))) float v2f;
typedef __attribute__((ext_vector_type(8))) float v8f;

#define HEADS 8
#define CH    32            // channels per head
#define CTOT  256
#define TT    8
#define HH    64
#define WW    64
#define SS    134           // t + h + w - 2
#define HW    (HH * WW)     // 4096
#define THW   (TT * HW)     // 32768
#define OFF_H TT            // A_h starts at s=8
#define OFF_W (TT + HH - 1) // A_w starts at s=71
#define LPAD  65            // LDS row stride (floats): 64+1 avoids bank conflicts

// ---------------------------------------------------------------------------
// Kernel 1: pv[o, x] = sum_c conv_w[o, c] * value[c, x] + conv_b[o]
// Wave tile: 16(M) x 64(N); K stepped by 4 with V_WMMA_F32_16X16X4_F32.
// A 16x4 layout: lanes 0-15 hold {K0,K1}, lanes 16-31 hold {K2,K3}.
// B 4x16 layout: mirror of A. C/D: VGPR r -> M=r / M=r+8, N=lane&15.
// ---------------------------------------------------------------------------
__global__ __launch_bounds__(256) void conv1x1_wmma(
    const float* __restrict__ Wm, const float* __restrict__ V,
    const float* __restrict__ bias, float* __restrict__ pv)
{
  const int lane  = threadIdx.x & 31;
  const int wave  = threadIdx.x >> 5;
  const int tile  = blockIdx.x * 8 + wave;  // 8192 wave tiles total
  const int mtile = tile >> 9;              // 0..15  (16 M-tiles of 16)
  const int ntile = tile & 511;             // 0..511 (512 N-tiles of 64)
  const int n0    = ntile * 64;
  const int l15   = lane & 15;
  const int half  = lane >> 4;
  const int mA    = mtile * 16 + l15;       // A-matrix row for this lane

  v8f acc[4] = {};

  for (int kk = 0; kk < CTOT; kk += 4) {
    const int k0 = kk + 2 * half;
    v2f a;
    a.x = Wm[mA * CTOT + k0];
    a.y = Wm[mA * CTOT + k0 + 1];
    const float* vrow = V + (size_t)k0 * THW;
#pragma unroll
    for (int j = 0; j < 4; ++j) {
      const int n = n0 + j * 16 + l15;
      v2f b;
      b.x = vrow[n];
      b.y = vrow[THW + n];
      acc[j] = __builtin_amdgcn_wmma_f32_16x16x4_f32(
          false, a, false, b, (short)0, acc[j], false, false);
    }
  }

#pragma unroll
  for (int r = 0; r < 8; ++r) {
    const int mrow = mtile * 16 + r + 8 * half;
    const float bb = bias[mrow];
#pragma unroll
    for (int j = 0; j < 4; ++j) {
      const int n = n0 + j * 16 + l15;
      pv[(size_t)mrow * THW + n] = acc[j][r] + bb;
    }
  }
}

// ---------------------------------------------------------------------------
// Kernel 2a: t-axis pass (K=8, too small for WMMA; scalar FMAs).
//   out[hd,c,t,h,w] = sum_s A[hd, s, t, h, w] * pv[hd, c, s, h, w]
// One thread per (hd, c, h, w); lanes vary along w -> coalesced A/pv/out.
// ---------------------------------------------------------------------------
__global__ __launch_bounds__(256) void attn_taxis(
    const float* __restrict__ A, const float* __restrict__ pv,
    float* __restrict__ out)
{
  const int idx = blockIdx.x * 256 + threadIdx.x;  // 2^20 threads
  const int w   = idx & 63;
  const int h   = (idx >> 6) & 63;
  const int c   = (idx >> 12) & 31;
  const int hd  = idx >> 17;
  const int xy  = h * WW + w;

  float v[TT];
#pragma unroll
  for (int s = 0; s < TT; ++s)
    v[s] = pv[((size_t)(hd * CH + c) * TT + s) * HW + xy];

#pragma unroll
  for (int t = 0; t < TT; ++t) {
    float accv = 0.f;
#pragma unroll
    for (int s = 0; s < TT; ++s)
      accv += A[((size_t)(hd * SS + s) * TT + t) * HW + xy] * v[s];
    out[((size_t)(hd * CH + c) * TT + t) * HW + xy] = accv;
  }
}

// ---------------------------------------------------------------------------
// Kernels 2b/2c: axial pass along h (AXIS=0) or w (AXIS=1).
// Per (head, t, fix): D[c=32, q=64] += V[c, p=64] @ Afull[p, q]
//   Afull[p,q] = 0 if p==q else Aslice[p<q ? p : p-1][q]
// Block = 4 waves / one GEMM. pv tile (32x64) and attention slice (63x64)
// are staged in LDS (row stride 65 -> conflict-free strided reads); each
// wave then runs the full K loop on a 16-wide q slab from LDS only.
// ---------------------------------------------------------------------------
template <int AXIS>
__global__ __launch_bounds__(128) void attn_axis_wmma(
    const float* __restrict__ A, const float* __restrict__ pv,
    float* __restrict__ out)
{
  __shared__ float Vt[CH * LPAD];        //  32 rows (c)   x 64 (p) -> 8.3 KB
  __shared__ float At[(HH - 1) * LPAD];  //  63 rows (s)   x 64 (q) -> 16.4 KB

  const int tid  = threadIdx.x;
  const int lane = tid & 31;
  const int wave = tid >> 5;
  const int fix  = blockIdx.x & 63;          // w if AXIS==0, h if AXIS==1
  const int t    = (blockIdx.x >> 6) & 7;
  const int hd   = blockIdx.x >> 9;

  const int l15  = lane & 15;
  const int half = lane >> 4;
  const int q    = wave * 16 + l15;          // output position along axis

  const int pstride  = (AXIS == 0) ? WW : 1; // pv/out stride along the axis
  const int axis_off = (AXIS == 0) ? OFF_H : OFF_W;
  const int fixoff   = (AXIS == 0) ? fix : fix * WW;

  // base for pv/out at c=0, axis position 0
  const size_t vbase = ((size_t)(hd * CH) * TT + t) * HW + fixoff;
  // base for this block's attention slice at s=0, q=0
  const size_t abase = ((size_t)(hd * SS + axis_off) * TT + t) * HW + fixoff;

  // ---- stage pv tile: Vt[c][p] = pv[hd, c, t, <p along axis>, fix] --------
#pragma unroll
  for (int i = 0; i < (CH * HH) / 128; ++i) {   // 16 iterations
    const int idx = i * 128 + tid;
    const int c   = idx >> 6;
    const int p   = idx & 63;
    Vt[c * LPAD + p] = pv[vbase + (size_t)c * THW + (size_t)p * pstride];
  }
  // ---- stage attention slice: At[s][q] = A[hd, off+s, t, <q along axis>] --
  for (int idx = tid; idx < (HH - 1) * HH; idx += 128) {  // 4032 elements
    const int s = idx >> 6;
    const int qq = idx & 63;
    At[s * LPAD + qq] = A[abase + (size_t)s * THW + (size_t)qq * pstride];
  }
  __syncthreads();

  // ---- WMMA K-loop, operands from LDS -------------------------------------
  v8f acc0 = {}, acc1 = {};
  for (int p0 = 0; p0 < HH; p0 += 4) {
    const int k0 = p0 + 2 * half;

    v2f a0, a1;
    a0.x = Vt[l15 * LPAD + k0];
    a0.y = Vt[l15 * LPAD + k0 + 1];
    a1.x = Vt[(l15 + 16) * LPAD + k0];
    a1.y = Vt[(l15 + 16) * LPAD + k0 + 1];

    v2f b;
#pragma unroll
    for (int e = 0; e < 2; ++e) {
      const int p = k0 + e;
      const int sidx = (p < q) ? p : (p - 1);  // in [0,62] whenever p != q
      float bv = 0.f;
      if (p != q) bv = At[sidx * LPAD + q];
      if (e == 0) b.x = bv; else b.y = bv;
    }

    acc0 = __builtin_amdgcn_wmma_f32_16x16x4_f32(
        false, a0, false, b, (short)0, acc0, false, false);
    acc1 = __builtin_amdgcn_wmma_f32_16x16x4_f32(
        false, a1, false, b, (short)0, acc1, false, false);
  }

  // ---- accumulate into out (unique element per lane within this kernel) ---
#pragma unroll
  for (int r = 0; r < 8; ++r) {
    const int crow = r + 8 * half;
    const size_t o0 = vbase + (size_t)crow * THW + (size_t)q * pstride;
    const size_t o1 = vbase + (size_t)(crow + 16) * THW + (size_t)q * pstride;
    out[o0] += acc0[r];
    out[o1] += acc1[r];
  }
}

// ---------------------------------------------------------------------------
extern "C" void kernel_launch(void* const* d_in, const int* in_sizes, int n_in,
                              void* d_out, int out_size, void* d_ws, size_t ws_size,
                              hipStream_t stream)
{
  (void)in_sizes; (void)n_in; (void)out_size; (void)ws_size;

  const float* attention = (const float*)d_in[0]; // [8*134, 8, 64, 64]
  const float* value     = (const float*)d_in[1]; // [256, 8, 64, 64]
  const float* conv_w    = (const float*)d_in[2]; // [256, 256]
  const float* conv_b    = (const float*)d_in[3]; // [256]
  float*       out       = (float*)d_out;         // [8, 32, 8, 64, 64]
  float*       pv        = (float*)d_ws;          // needs 256*32768*4 = 32 MiB

  // 1) 1x1 conv GEMM: 16 M-tiles * 512 N-tiles = 8192 waves / 8 per block
  conv1x1_wmma<<<1024, 256, 0, stream>>>(conv_w, value, conv_b, pv);

  // 2a) t-axis pass (writes out)
  attn_taxis<<<4096, 256, 0, stream>>>(attention, pv, out);

  // 2b) h-axis pass, 2c) w-axis pass (accumulate into out)
  attn_axis_wmma<0><<<HEADS * TT * WW, 128, 0, stream>>>(attention, pv, out);
  attn_axis_wmma<1><<<HEADS * TT * HH, 128, 0, stream>>>(attention, pv, out);
}